// VQEModule_16028817948750
// MI455X (gfx1250) — compile-verified
//
#include <hip/hip_runtime.h>
#include <math.h>

// ---- problem constants (N=8 qubits) ----
#define NQ     8
#define DIMN   256
#define NLAYER 256

typedef float v2f __attribute__((ext_vector_type(2)));
typedef float v8f __attribute__((ext_vector_type(8)));

__device__ __forceinline__ float2 cmul(float2 a, float2 b) {
    return make_float2(a.x * b.x - a.y * b.y, a.x * b.y + a.y * b.x);
}
__device__ __forceinline__ float2 cadd(float2 a, float2 b) {
    return make_float2(a.x + b.x, a.y + b.y);
}

__global__ __launch_bounds__(256, 1)
void vqe_scan_obs_kernel(const float* __restrict__ params, // (256, 8, 3)
                         const float* __restrict__ phi,    // (256, 7)
                         const float* __restrict__ Hre,    // (256, 256)
                         const float* __restrict__ Him,    // (256, 256)
                         float* __restrict__ out)          // scalar
{
    // Gate coefficient tables for ALL layers (computed once, in parallel).
    __shared__ float2 gU[NLAYER][NQ][4];      // 64 KB : U3 entries u00,u01,u10,u11
    __shared__ float2 gPM[NLAYER][NQ - 1][2]; // 28 KB : entangler p, m
    __shared__ float2 xbuf[2][DIMN];          //  4 KB : ping-pong cross-wave exchange
    __shared__ float2 sfin[DIMN];             //  2 KB : final state
    __shared__ float  dRe[2][DIMN];           //  2 KB : (Hre @ [sre|sim]) cols 0,1
    __shared__ float  dIm[2][DIMN];           //  2 KB : (Him^T @ [sre|sim]) cols 0,1
    __shared__ float  wsum[8];

    const int t = threadIdx.x;

    // ------------------------------------------------------------------
    // Phase A: thread l computes ALL gate coefficients for layer l.
    // u3: row0 = [ct, -e^{i lam} st], row1 = [e^{i phi} st, e^{i(phi+lam)} ct]
    // entangler: e = exp(4i phi_j), p = (1+e)/2, m = (1-e)/2
    // ------------------------------------------------------------------
    {
        const int l = t;
        #pragma unroll
        for (int i = 0; i < NQ; ++i) {
            const float th = params[(l * NQ + i) * 3 + 0];
            const float ph = params[(l * NQ + i) * 3 + 1];
            const float la = params[(l * NQ + i) * 3 + 2];
            float st, ct, sl, cl, sp, cp, spl, cpl;
            sincosf(0.5f * th, &st, &ct);
            sincosf(la, &sl, &cl);
            sincosf(ph, &sp, &cp);
            sincosf(ph + la, &spl, &cpl);
            gU[l][i][0] = make_float2(ct, 0.0f);
            gU[l][i][1] = make_float2(-cl * st, -sl * st);
            gU[l][i][2] = make_float2(cp * st, sp * st);
            gU[l][i][3] = make_float2(cpl * ct, spl * ct);
        }
        #pragma unroll
        for (int j = 0; j < NQ - 1; ++j) {
            const float a = 4.0f * phi[l * (NQ - 1) + j];
            float sa, ca;
            sincosf(a, &sa, &ca);
            gPM[l][j][0] = make_float2(0.5f * (1.0f + ca),  0.5f * sa); // p
            gPM[l][j][1] = make_float2(0.5f * (1.0f - ca), -0.5f * sa); // m
        }
    }
    __syncthreads();

    // ------------------------------------------------------------------
    // Phase B: sequential scan. Amplitude index == threadIdx.x.
    // bits [4:0] = lane (intra-wave via shfl_xor), bits [7:5] = wave
    // (cross-wave via LDS ping-pong, 1 barrier per cross gate).
    // Qubit i <-> bit (7 - i).  Entanglers applied K6..K0 (reference order).
    // ------------------------------------------------------------------
    float2 amp = make_float2(t == 0 ? 1.0f : 0.0f, 0.0f);
    int pp = 0;

    for (int l = 0; l < NLAYER; ++l) {
        // --- rotation layer: 8 single-qubit U3 gates (they commute) ---
        #pragma unroll
        for (int i = 0; i < NQ; ++i) {
            const int b = 7 - i;
            float2 other;
            if (b >= 5) {
                xbuf[pp][t] = amp;
                __syncthreads();
                other = xbuf[pp][t ^ (1 << b)];
                pp ^= 1;
            } else {
                other.x = __shfl_xor(amp.x, 1 << b, 32);
                other.y = __shfl_xor(amp.y, 1 << b, 32);
            }
            const int bv = (t >> b) & 1;
            const float2 x0 = bv ? other : amp;   // partner with bit==0
            const float2 x1 = bv ? amp : other;   // partner with bit==1
            const float2 e0 = gU[l][i][2 * bv + 0];
            const float2 e1 = gU[l][i][2 * bv + 1];
            amp = cadd(cmul(e0, x0), cmul(e1, x1));
        }
        // --- entangling layer: full = K0@K1@...@K6 => apply K6 first ---
        #pragma unroll
        for (int j = NQ - 2; j >= 0; --j) {
            const int tb = 6 - j;          // target qubit j+1 -> bit 6-j
            float2 other;
            if (tb >= 5) {
                xbuf[pp][t] = amp;
                __syncthreads();
                other = xbuf[pp][t ^ (1 << tb)];
                pp ^= 1;
            } else {
                other.x = __shfl_xor(amp.x, 1 << tb, 32);
                other.y = __shfl_xor(amp.y, 1 << tb, 32);
            }
            const int cb = (t >> (7 - j)) & 1; // control qubit j -> bit 7-j
            const float2 p = gPM[l][j][0];
            const float2 m = gPM[l][j][1];
            const float2 mix = cadd(cmul(p, amp), cmul(m, other));
            const float2 nv = cb ? mix : amp;
            amp = make_float2(-nv.x, -nv.y);   // global -1 of the noisy CNOT
        }
    }

    sfin[t] = amp;
    __syncthreads();

    // ------------------------------------------------------------------
    // Phase C: obs = Re(s^H H s) with f32 WMMA (V_WMMA_F32_16X16X4_F32).
    // t_re = Hre@sre - Him@sim,  t_im = Hre@sim + Him@sre,
    // obs = sum_r sre_r*t_re_r + sim_r*t_im_r.
    // Hermitian symmetry => coalesced A loads from transposed orientation:
    //   Hre[r][c] = Hre[c*256+r] ; Him[r][c] = -Him[c*256+r] (sign folded below).
    // Wave w owns rows [32w, 32w+32): two 16-row WMMA tiles.
    // A layout (16x4 f32): lanes0-15 M=0..15 {VGPR0=K0,VGPR1=K1};
    //                      lanes16-31        {VGPR0=K2,VGPR1=K3}.
    // B (4x16): VGPR0 = row K{0|2} striped over lane halves, VGPR1 = K{1|3}.
    // B col 0 = sre chunk, col 1 = sim chunk, cols 2..15 = 0.
    // ------------------------------------------------------------------
    const int lane = t & 31;
    const int wv   = t >> 5;
    const int half = lane >> 4;
    const int m    = lane & 15;
    const int R0   = wv * 32;
    const int R1   = R0 + 16;

    v8f accRe0 = {0.f, 0.f, 0.f, 0.f, 0.f, 0.f, 0.f, 0.f};
    v8f accIm0 = accRe0, accRe1 = accRe0, accIm1 = accRe0;

    for (int k = 0; k < DIMN; k += 4) {
        const int c0 = k + 2 * half;
        const int c1 = c0 + 1;

        const float2 s0 = sfin[c0];
        const float2 s1 = sfin[c1];
        v2f b;
        b.x = (m == 0) ? s0.x : (m == 1) ? s0.y : 0.0f;
        b.y = (m == 0) ? s1.x : (m == 1) ? s1.y : 0.0f;

        v2f aRe0v, aIm0v, aRe1v, aIm1v;
        aRe0v.x = Hre[c0 * DIMN + R0 + m];  aRe0v.y = Hre[c1 * DIMN + R0 + m];
        aIm0v.x = Him[c0 * DIMN + R0 + m];  aIm0v.y = Him[c1 * DIMN + R0 + m];
        aRe1v.x = Hre[c0 * DIMN + R1 + m];  aRe1v.y = Hre[c1 * DIMN + R1 + m];
        aIm1v.x = Him[c0 * DIMN + R1 + m];  aIm1v.y = Him[c1 * DIMN + R1 + m];

        accRe0 = __builtin_amdgcn_wmma_f32_16x16x4_f32(false, aRe0v, false, b,
                                                       (short)0, accRe0, false, false);
        accIm0 = __builtin_amdgcn_wmma_f32_16x16x4_f32(false, aIm0v, false, b,
                                                       (short)0, accIm0, false, false);
        accRe1 = __builtin_amdgcn_wmma_f32_16x16x4_f32(false, aRe1v, false, b,
                                                       (short)0, accRe1, false, false);
        accIm1 = __builtin_amdgcn_wmma_f32_16x16x4_f32(false, aIm1v, false, b,
                                                       (short)0, accIm1, false, false);
    }

    // C/D layout: VGPR v -> row (v + 8*half), col = lane&15. Need cols 0,1 only.
    if (m < 2) {
        #pragma unroll
        for (int v = 0; v < 8; ++v) {
            const int r0 = R0 + v + 8 * half;
            const int r1 = R1 + v + 8 * half;
            dRe[m][r0] = accRe0[v];
            dIm[m][r0] = accIm0[v];
            dRe[m][r1] = accRe1[v];
            dIm[m][r1] = accIm1[v];
        }
    }
    __syncthreads();

    // dRe[c][r] = (Hre @ [sre|sim])[r,c];  dIm[c][r] = -(Him @ [sre|sim])[r,c]
    const float tre = dRe[0][t] + dIm[1][t];
    const float tim = dRe[1][t] - dIm[0][t];
    float partial = sfin[t].x * tre + sfin[t].y * tim;

    #pragma unroll
    for (int o = 16; o > 0; o >>= 1)
        partial += __shfl_xor(partial, o, 32);
    if (lane == 0) wsum[wv] = partial;
    __syncthreads();

    if (t == 0) {
        float s = 0.0f;
        #pragma unroll
        for (int w = 0; w < 8; ++w) s += wsum[w];
        out[0] = s;
    }
}

extern "C" void kernel_launch(void* const* d_in, const int* in_sizes, int n_in,
                              void* d_out, int out_size, void* d_ws, size_t ws_size,
                              hipStream_t stream) {
    const float* params = (const float*)d_in[0]; // (256, 8, 3)
    const float* phi    = (const float*)d_in[1]; // (256, 7)
    const float* Hre    = (const float*)d_in[2]; // (256, 256)
    const float* Him    = (const float*)d_in[3]; // (256, 256)
    float* out = (float*)d_out;

    (void)in_sizes; (void)n_in; (void)d_ws; (void)ws_size; (void)out_size;

    // Sequential scan => single workgroup, 256 threads (8 wave32s on one WGP).
    vqe_scan_obs_kernel<<<1, 256, 0, stream>>>(params, phi, Hre, Him, out);
}